// FlexibleLoss_40553081209497
// MI455X (gfx1250) — compile-verified
//
#include <hip/hip_runtime.h>

// ---------------------------------------------------------------------------
// FlexibleLoss forward for MI455X (gfx1250, wave32).
//  Kernel 1: one 256-thread block (8 waves) per row b of [B, K, 2] logits.
//            Streams logits once from HBM (bandwidth floor ~1.4us @23.3TB/s),
//            stages d = l0-l1 in LDS, does 5 block reductions where the
//            wave32 step is V_WMMA_F32_16X16X4_F32 (A=values, B=ones).
//  Kernel 2: deterministic final sum of per-row contributions (no atomics).
// ---------------------------------------------------------------------------

typedef __attribute__((ext_vector_type(2))) float v2f;
typedef __attribute__((ext_vector_type(8))) float v8f;

#define THREADS 256
#define NWAVES  (THREADS / 32)
#define CE_T    5
#define EPSF    1e-6f
#define MAXK    2048   // K = 1000 for this problem; LDS row buffer

// Full wave32 sum via one f32 WMMA: A[m,0]=v[lane m], A[m,2]=v[lane 16+m],
// B = ones(4x16)  =>  D[m,n] = v[m] + v[m+16] for every n.
// Lanes 0-15 hold D rows 0..7 in d[0..7]; lanes 16-31 hold rows 8..15.
// 7 adds + one half-swap shuffle yields the exact-fp32 wave sum in all lanes.
__device__ __forceinline__ float wave_reduce_wmma(float v) {
  v2f a;   a.x = v;    a.y = 0.0f;
  v2f one; one.x = 1.0f; one.y = 1.0f;
  v8f c = {};
  v8f d = __builtin_amdgcn_wmma_f32_16x16x4_f32(
      /*neg_a=*/false, a, /*neg_b=*/false, one,
      /*c_mod=*/(short)0, c, /*reuse_a=*/false, /*reuse_b=*/false);
  float s = ((d[0] + d[1]) + (d[2] + d[3])) + ((d[4] + d[5]) + (d[6] + d[7]));
  s += __shfl_xor(s, 16, 32);
  return s;
}

// Block-wide sum, identical value in every lane; fixed combine order so the
// result is bit-deterministic across replays. Must be called convergently.
__device__ __forceinline__ float block_reduce(float v, int tid) {
  __shared__ float s_part[NWAVES];
  float w = wave_reduce_wmma(v);
  __syncthreads();                       // protect s_part reuse across calls
  if ((tid & 31) == 0) s_part[tid >> 5] = w;
  __syncthreads();
  float t = 0.0f;
#pragma unroll
  for (int k = 0; k < NWAVES; ++k) t += s_part[k];
  return t;
}

__global__ void __launch_bounds__(THREADS)
flexible_loss_row_kernel(const float* __restrict__ logits,
                         const int*   __restrict__ labels,
                         const int*   __restrict__ epoch_num,
                         float*       __restrict__ row_out,
                         int B, int K) {
  __shared__ float ds_d[MAXK];
  __shared__ float s_bl0lab;

  const int b   = blockIdx.x;
  const int tid = threadIdx.x;
  const int lab = labels[b];
  const float2* row = reinterpret_cast<const float2*>(logits) + (size_t)b * K;

  // ---- pass 1: stream logits once; 2-way log-softmax per class -----------
  float acc_bl1 = 0.0f;   // sum_i log p1  -> ignorance_log
  float acc_ce  = 0.0f;   // sum_i ce_per[b, i]
  for (int i = tid; i < K; i += THREADS) {
    float2 v = row[i];
    float l0 = v.x, l1 = v.y;
    float m   = fmaxf(l0, l1);
    float lse = m + __logf(__expf(l0 - m) + __expf(l1 - m));
    float bl0 = l0 - lse;
    float bl1 = l1 - lse;
    ds_d[i]   = l0 - l1;           // bel_term_log
    acc_bl1  += bl1;
    if (i == lab) { acc_ce -= bl0; s_bl0lab = bl0; }
    else          { acc_ce -= bl1; }
  }
  float ig     = block_reduce(acc_bl1, tid);   // ignorance_log[b]
  float ce_row = block_reduce(acc_ce,  tid);

  // ---- pass 2a: belief = exp(ig + d) (matches reference underflow) -------
  float acc_bel = 0.0f;
  for (int i = tid; i < K; i += THREADS) {
    float bel = __expf(ig + ds_d[i]);
    ds_d[i]   = bel;
    acc_bel  += bel;
  }
  float sum_bel = block_reduce(acc_bel, tid);
  float u = 1.0f - sum_bel;                    // uncertainty
  float S = (float)K / (u + EPSF);

  // ---- pass 2b: alpha = belief*S + 1 ; S_a = sum alpha --------------------
  float acc_alpha = 0.0f;
  for (int i = tid; i < K; i += THREADS) {
    float alpha = ds_d[i] * S + 1.0f;
    ds_d[i]     = alpha;
    acc_alpha  += alpha;
  }
  float S_a = block_reduce(acc_alpha, tid);
  float den = S_a * S_a * (S_a + 1.0f);

  // ---- pass 2c: err + var -------------------------------------------------
  float acc_ev = 0.0f;
  for (int i = tid; i < K; i += THREADS) {
    float alpha = ds_d[i];
    float p  = alpha / S_a;
    float yi = (i == lab) ? 1.0f : 0.0f;
    float t  = yi - p;
    acc_ev  += t * t + alpha * (S_a - alpha) / den;
  }
  float edl1_row = block_reduce(acc_ev, tid);

  if (tid == 0) {
    float plaus_max = 1.0f - __expf(ig);
    float plaus_lab = __expf(s_bl0lab);
    float d2   = plaus_lab - plaus_max;
    float edl2 = d2 * d2;
    float ce_flag = (epoch_num[0] < CE_T) ? 1.0f : 0.0f;
    float inv_B  = 1.0f / (float)B;
    float inv_BK = inv_B / (float)K;
    row_out[b] = ce_flag * ce_row * inv_BK + (edl1_row + edl2) * inv_B;
  }
}

__global__ void __launch_bounds__(THREADS)
flexible_loss_final_reduce(const float* __restrict__ row_out,
                           float*       __restrict__ out, int B) {
  const int tid = threadIdx.x;
  float acc = 0.0f;
  for (int r = tid; r < B; r += THREADS) acc += row_out[r];  // fixed order
  float total = block_reduce(acc, tid);
  if (tid == 0) out[0] = total;
}

extern "C" void kernel_launch(void* const* d_in, const int* in_sizes, int n_in,
                              void* d_out, int out_size, void* d_ws, size_t ws_size,
                              hipStream_t stream) {
  const float* logits = (const float*)d_in[0];
  const int*   labels = (const int*)d_in[1];
  const int*   epoch  = (const int*)d_in[2];

  const int B = in_sizes[1];                // 4096
  const int K = in_sizes[0] / (B * 2);      // 1000

  float* row_ws = (float*)d_ws;             // B floats of scratch

  flexible_loss_row_kernel<<<B, THREADS, 0, stream>>>(
      logits, labels, epoch, row_ws, B, K);
  flexible_loss_final_reduce<<<1, THREADS, 0, stream>>>(
      row_ws, (float*)d_out, B);
}